// StandardizationLayer_81277961109766
// MI455X (gfx1250) — compile-verified
//
#include <hip/hip_runtime.h>

// Per-image standardization of a (64, 512, 512, 3) fp32 tensor on gfx1250.
// Pass 1: per-(image,chunk) partial sum / sum-of-squares, wave reduction via
//         V_WMMA_F32_16X16X4_F32 (all-ones B), block reduction via LDS.
// Pass 2: one wave per image folds 32 partials, computes mean and 1/adjusted_std.
// Pass 3: elementwise normalize; NT loads (input is dead after this) and NT
//         stores (don't let the output stream evict input lines from the
//         192 MB L2 that pass 3 wants to re-hit).

typedef float v2f __attribute__((ext_vector_type(2)));
typedef float v4f __attribute__((ext_vector_type(4)));
typedef float v8f __attribute__((ext_vector_type(8)));

#define N_IMG          64
#define IMG_ELEMS      (512 * 512 * 3)        // 786432 floats / image
#define IMG_VEC4       (IMG_ELEMS / 4)        // 196608 float4 / image
#define P_BLOCKS       32                     // chunks per image
#define THREADS        256                    // 8 waves (wave32)
#define VEC4_PER_BLOCK (IMG_VEC4 / P_BLOCKS)  // 6144 float4 / chunk
#define WAVES          (THREADS / 32)

// Reduce one fp32 value across all 32 lanes of a wave using
// V_WMMA_F32_16X16X4_F32 with B = ones (layout-independent for B).
// A (16x4): lanes 0-15 hold M=0..15 with K=0 in VGPR0, K=1 in VGPR1;
// lanes 16-31 hold K=2,3.  With A0=x, A1=0:  D[m][n] = x(m) + x(m+16).
// Lane n (0-15) holds D[0..7][n] in d[0..7]; lane n+16 holds D[8..15][n].
// Sum the 8 accumulator VGPRs, then one xor-16 shuffle -> full 32-lane sum.
__device__ __forceinline__ float wave_sum_wmma(float x) {
  v2f a; a[0] = x;    a[1] = 0.0f;
  v2f b; b[0] = 1.0f; b[1] = 1.0f;
  v8f c = {};
  v8f d = __builtin_amdgcn_wmma_f32_16x16x4_f32(
      /*neg_a=*/false, a, /*neg_b=*/false, b,
      /*c_mod=*/(short)0, c, /*reuse_a=*/false, /*reuse_b=*/false);
  float v = ((d[0] + d[1]) + (d[2] + d[3])) + ((d[4] + d[5]) + (d[6] + d[7]));
  v += __shfl_xor(v, 16, 32);
  return v;  // every lane holds the total
}

__global__ __launch_bounds__(THREADS) void std_partial_reduce(
    const float* __restrict__ in, float2* __restrict__ partial) {
  const int img = blockIdx.y;
  const int blk = blockIdx.x;
  const v4f* __restrict__ src = (const v4f*)in +
      (size_t)img * IMG_VEC4 + (size_t)blk * VEC4_PER_BLOCK;

  float s = 0.0f, q = 0.0f;
  // 6144 / 256 = 24 iterations, identical for all lanes -> EXEC stays all-1s.
  for (int i = threadIdx.x; i < VEC4_PER_BLOCK; i += THREADS) {
    v4f v = src[i];
    s += (v[0] + v[1]) + (v[2] + v[3]);
    q = fmaf(v[0], v[0], q);
    q = fmaf(v[1], v[1], q);
    q = fmaf(v[2], v[2], q);
    q = fmaf(v[3], v[3], q);
  }

  const float ws = wave_sum_wmma(s);
  const float wq = wave_sum_wmma(q);

  __shared__ float2 lds[WAVES];
  const int wave = threadIdx.x >> 5;
  const int lane = threadIdx.x & 31;
  if (lane == 0) lds[wave] = make_float2(ws, wq);
  __syncthreads();
  if (threadIdx.x == 0) {
    float S = 0.0f, Q = 0.0f;
    #pragma unroll
    for (int w = 0; w < WAVES; ++w) { S += lds[w].x; Q += lds[w].y; }
    partial[img * P_BLOCKS + blk] = make_float2(S, Q);
  }
}

__global__ __launch_bounds__(32) void std_finalize_stats(
    const float2* __restrict__ partial,
    float* __restrict__ meanArr, float* __restrict__ invArr) {
  const int img  = blockIdx.x;
  const int lane = threadIdx.x;           // P_BLOCKS == 32 == one wave
  float2 p = partial[img * P_BLOCKS + lane];
  float S = p.x, Q = p.y;
  #pragma unroll
  for (int off = 16; off >= 1; off >>= 1) {  // deterministic fixed-order tree
    S += __shfl_xor(S, off, 32);
    Q += __shfl_xor(Q, off, 32);
  }
  if (lane == 0) {
    const float n    = (float)IMG_ELEMS;
    const float mean = S / n;
    const float var  = Q / n - mean * mean;
    const float sd   = sqrtf(fmaxf(var, 0.0f));
    const float adj  = fmaxf(sd, 1.0f / sqrtf(n));  // TF stddev clamp 1/sqrt(N)
    meanArr[img] = mean;
    invArr[img]  = 1.0f / adj;
  }
}

__global__ __launch_bounds__(THREADS) void std_normalize(
    const float* __restrict__ in, float* __restrict__ out,
    const float* __restrict__ meanArr, const float* __restrict__ invArr) {
  const int img = blockIdx.y;
  const int blk = blockIdx.x;
  const float mean = meanArr[img];
  const float inv  = invArr[img];
  const size_t base = (size_t)img * IMG_VEC4 + (size_t)blk * VEC4_PER_BLOCK;
  const v4f* __restrict__ src = (const v4f*)in + base;
  v4f* __restrict__ dst = (v4f*)out + base;

  for (int i = threadIdx.x; i < VEC4_PER_BLOCK; i += THREADS) {
    v4f v = __builtin_nontemporal_load(&src[i]);   // last use of input
    v4f r;
    r[0] = (v[0] - mean) * inv;
    r[1] = (v[1] - mean) * inv;
    r[2] = (v[2] - mean) * inv;
    r[3] = (v[3] - mean) * inv;
    __builtin_nontemporal_store(r, &dst[i]);       // don't pollute L2
  }
}

extern "C" void kernel_launch(void* const* d_in, const int* in_sizes, int n_in,
                              void* d_out, int out_size, void* d_ws, size_t ws_size,
                              hipStream_t stream) {
  (void)in_sizes; (void)n_in; (void)out_size; (void)ws_size;

  const float* in  = (const float*)d_in[0];
  float*       out = (float*)d_out;

  // Workspace: 64*32 float2 partials (16 KB) + 64 means + 64 inv-stddevs.
  float2* partial = (float2*)d_ws;
  float*  meanArr = (float*)d_ws + 2 * N_IMG * P_BLOCKS;
  float*  invArr  = meanArr + N_IMG;

  dim3 grid(P_BLOCKS, N_IMG);  // same geometry pass 1 & 3 -> L2 temporal reuse
  std_partial_reduce<<<grid, THREADS, 0, stream>>>(in, partial);
  std_finalize_stats<<<N_IMG, 32, 0, stream>>>(partial, meanArr, invArr);
  std_normalize<<<grid, THREADS, 0, stream>>>(in, out, meanArr, invArr);
}